// FeaSAttack_42021960024106
// MI455X (gfx1250) — compile-verified
//
#include <hip/hip_runtime.h>
#include <hip/hip_bf16.h>
#include <math.h>

#define N        8192
#define DFEAT    512
#define EPS_SINK 0.01f
#define INV_EPS  100.0f
#define THRESH   0.1f
#define LSE_EPS  1e-6f
#define COS_EPS  1e-6f
#define LOG_MU   (-9.010913347f)   /* -log(8192) */
#define LOG_NU   (-9.010913347f)
#define NITER    100
#define FINAL_BLOCKS 4096

typedef __attribute__((ext_vector_type(2))) float v2f;
typedef __attribute__((ext_vector_type(8))) float v8f;

// ---------------- block reduction (wave32, 256 threads) ----------------
__device__ __forceinline__ float blockReduce256(float v) {
    __shared__ float s[8];
    #pragma unroll
    for (int o = 16; o > 0; o >>= 1) v += __shfl_xor(v, o, 32);
    if ((threadIdx.x & 31) == 0) s[threadIdx.x >> 5] = v;
    __syncthreads();
    if (threadIdx.x < 32) {
        v = (threadIdx.x < 8) ? s[threadIdx.x] : 0.0f;
        #pragma unroll
        for (int o = 4; o > 0; o >>= 1) v += __shfl_xor(v, o, 32);
    }
    return v;   // valid on thread 0
}

// ---------------- row norms of X and Y ----------------
__global__ __launch_bounds__(256) void norms_kernel(const float* __restrict__ X,
                                                    const float* __restrict__ Y,
                                                    float* __restrict__ xn,
                                                    float* __restrict__ yn) {
    int b = blockIdx.x;
    const float* row;
    float* dst;
    int r;
    if (b < N) { row = X + (size_t)b * DFEAT; dst = xn; r = b; }
    else       { row = Y + (size_t)(b - N) * DFEAT; dst = yn; r = b - N; }
    float s = 0.0f;
    for (int k = threadIdx.x; k < DFEAT; k += 256) { float t = row[k]; s += t * t; }
    s = blockReduce256(s);
    if (threadIdx.x == 0) dst[r] = sqrtf(s);
}

// ---------------- init state ----------------
__global__ __launch_bounds__(256) void init_kernel(float* __restrict__ u,
                                                   float* __restrict__ v,
                                                   int* __restrict__ done) {
    int i = blockIdx.x * 256 + threadIdx.x;
    if (i < N) { u[i] = 0.0f; v[i] = 0.0f; }
    if (i == 0) { done[0] = 0; done[1] = 0; }
}

// ---------------- cost matrix via F32 WMMA ----------------
// One wave per 16x16 output tile. A = X (row-major, 16xK), B = Y^T (Kx16).
// Fragment layouts per CDNA5 ISA 7.12.2 (f32 16x4 A: lanes0-15 K=0/1 in v0/v1,
// lanes16-31 K=2/3; B 4x16 symmetric; D: vgpr r -> M=r (lanes0-15) / M=8+r).
__global__ __launch_bounds__(32) void cost_wmma_kernel(const float* __restrict__ X,
                                                       const float* __restrict__ Y,
                                                       const float* __restrict__ xn,
                                                       const float* __restrict__ yn,
                                                       float* __restrict__ C) {
    const int lane = threadIdx.x;
    const int half = lane >> 4;      // 0 or 1
    const int l    = lane & 15;
    const int row0 = blockIdx.y * 16;
    const int col0 = blockIdx.x * 16;
    const float* arow = X + (size_t)(row0 + l) * DFEAT;
    const float* brow = Y + (size_t)(col0 + l) * DFEAT;
    const int ko = half ? 2 : 0;

    v8f acc = {};
    for (int k = 0; k < DFEAT; k += 4) {
        v2f a, b;
        a.x = arow[k + ko];
        a.y = arow[k + ko + 1];
        b.x = brow[k + ko];
        b.y = brow[k + ko + 1];
        acc = __builtin_amdgcn_wmma_f32_16x16x4_f32(
            /*neg_a=*/false, a, /*neg_b=*/false, b,
            /*c_mod=*/(short)0, acc, /*reuse_a=*/false, /*reuse_b=*/false);
    }

    const int nn  = col0 + l;
    const float ynn = yn[nn];
    #pragma unroll
    for (int r = 0; r < 8; ++r) {
        int m = row0 + r + half * 8;
        float dot   = acc[r];
        float denom = fmaxf(xn[m] * ynn, COS_EPS);
        float cst   = fmaxf(1.0f - dot / denom, 0.0f);
        C[(size_t)m * N + nn] = cst;
    }
}

// ---------------- row pass: u update ----------------
__global__ __launch_bounds__(256) void row_kernel(const float* __restrict__ C,
                                                  float* __restrict__ u,
                                                  const float* __restrict__ v,
                                                  float* __restrict__ absd,
                                                  const int* __restrict__ done_r) {
    if (*done_r) return;
    const int i = blockIdx.x;
    const float ui = u[i];
    const float* crow = C + (size_t)i * N;
    float s = 0.0f;
    for (int j = threadIdx.x * 4; j < N; j += 256 * 4) {
        float4 c  = *(const float4*)(crow + j);
        float4 vj = *(const float4*)(v + j);
        s += __expf((ui + vj.x - c.x) * INV_EPS);
        s += __expf((ui + vj.y - c.y) * INV_EPS);
        s += __expf((ui + vj.z - c.z) * INV_EPS);
        s += __expf((ui + vj.w - c.w) * INV_EPS);
    }
    s = blockReduce256(s);
    if (threadIdx.x == 0) {
        float lse = __logf(s + LSE_EPS);
        float un  = EPS_SINK * (LOG_MU - lse) + ui;
        absd[i]   = fabsf(un - ui);
        u[i]      = un;
    }
}

// ---------------- col pass part 1: partial column sums ----------------
// grid (8, 64): 8 blocks x 256 thr x 4 cols = 8192 cols; 64 chunks x 128 rows.
__global__ __launch_bounds__(256) void colpart_kernel(const float* __restrict__ C,
                                                      const float* __restrict__ u,
                                                      const float* __restrict__ v,
                                                      float* __restrict__ part,
                                                      const int* __restrict__ done_r) {
    if (*done_r) return;
    const int j  = (blockIdx.x * 256 + threadIdx.x) * 4;
    const int r0 = blockIdx.y * 128;
    float4 vj = *(const float4*)(v + j);
    float4 s  = make_float4(0.f, 0.f, 0.f, 0.f);
    for (int i = 0; i < 128; ++i) {
        int row = r0 + i;
        float ui = u[row];
        float4 c = *(const float4*)(C + (size_t)row * N + j);
        s.x += __expf((ui + vj.x - c.x) * INV_EPS);
        s.y += __expf((ui + vj.y - c.y) * INV_EPS);
        s.z += __expf((ui + vj.z - c.z) * INV_EPS);
        s.w += __expf((ui + vj.w - c.w) * INV_EPS);
    }
    *(float4*)(part + (size_t)blockIdx.y * N + j) = s;
}

// ---------------- col pass part 2: v update + convergence flag ----------------
// blocks 0..7: v update for 1024 cols each; block 8: reduce |du|, write done.
__global__ __launch_bounds__(256) void colreduce_kernel(const float* __restrict__ part,
                                                        float* __restrict__ v,
                                                        const float* __restrict__ absd,
                                                        const int* __restrict__ done_r,
                                                        int* __restrict__ done_w) {
    if (blockIdx.x == 8) {
        float e = 0.0f;
        for (int i = threadIdx.x; i < N; i += 256) e += absd[i];
        e = blockReduce256(e);
        if (threadIdx.x == 0) *done_w = (*done_r) | (e < THRESH ? 1 : 0);
        return;
    }
    if (*done_r) return;
    const int j = (blockIdx.x * 256 + threadIdx.x) * 4;
    float4 s = make_float4(0.f, 0.f, 0.f, 0.f);
    for (int c = 0; c < 64; ++c) {
        float4 p = *(const float4*)(part + (size_t)c * N + j);
        s.x += p.x; s.y += p.y; s.z += p.z; s.w += p.w;
    }
    float4 vj = *(const float4*)(v + j);
    vj.x = EPS_SINK * (LOG_NU - __logf(s.x + LSE_EPS)) + vj.x;
    vj.y = EPS_SINK * (LOG_NU - __logf(s.y + LSE_EPS)) + vj.y;
    vj.z = EPS_SINK * (LOG_NU - __logf(s.z + LSE_EPS)) + vj.z;
    vj.w = EPS_SINK * (LOG_NU - __logf(s.w + LSE_EPS)) + vj.w;
    *(float4*)(v + j) = vj;
}

// ---------------- final: sum(T * C), two-phase deterministic reduction ----------
__global__ __launch_bounds__(256) void final_part_kernel(const float* __restrict__ C,
                                                         const float* __restrict__ u,
                                                         const float* __restrict__ v,
                                                         float* __restrict__ partials) {
    const size_t total4 = (size_t)N * N / 4;
    const size_t stride = (size_t)FINAL_BLOCKS * 256;
    float s = 0.0f;
    for (size_t q = (size_t)blockIdx.x * 256 + threadIdx.x; q < total4; q += stride) {
        size_t e0 = q * 4;
        int i = (int)(e0 >> 13);       // /8192
        int j = (int)(e0 & (N - 1));   // %8192 (float4 never crosses a row)
        float ui  = u[i];
        float4 c  = *(const float4*)(C + e0);
        float4 vj = *(const float4*)(v + j);
        s += __expf((ui + vj.x - c.x) * INV_EPS) * c.x;
        s += __expf((ui + vj.y - c.y) * INV_EPS) * c.y;
        s += __expf((ui + vj.z - c.z) * INV_EPS) * c.z;
        s += __expf((ui + vj.w - c.w) * INV_EPS) * c.w;
    }
    s = blockReduce256(s);
    if (threadIdx.x == 0) partials[blockIdx.x] = s;
}

__global__ __launch_bounds__(256) void final_reduce_kernel(const float* __restrict__ partials,
                                                           float* __restrict__ out) {
    float s = 0.0f;
    for (int i = threadIdx.x; i < FINAL_BLOCKS; i += 256) s += partials[i];
    s = blockReduce256(s);
    if (threadIdx.x == 0) out[0] = s;
}

// ---------------- host side ----------------
extern "C" void kernel_launch(void* const* d_in, const int* in_sizes, int n_in,
                              void* d_out, int out_size, void* d_ws, size_t ws_size,
                              hipStream_t stream) {
    const float* X = (const float*)d_in[0];
    const float* Y = (const float*)d_in[1];
    float* out = (float*)d_out;

    // workspace layout
    float* C     = (float*)d_ws;                 // N*N
    float* u     = C + (size_t)N * N;            // N
    float* v     = u + N;                        // N
    float* absd  = v + N;                        // N
    float* xn    = absd + N;                     // N
    float* yn    = xn + N;                       // N
    float* part  = yn + N;                       // 64*N
    float* fpart = part + (size_t)64 * N;        // FINAL_BLOCKS
    int*   done  = (int*)(fpart + FINAL_BLOCKS); // 2 ints (ping-pong)

    norms_kernel<<<2 * N, 256, 0, stream>>>(X, Y, xn, yn);
    init_kernel<<<N / 256, 256, 0, stream>>>(u, v, done);
    cost_wmma_kernel<<<dim3(N / 16, N / 16), 32, 0, stream>>>(X, Y, xn, yn, C);

    for (int it = 0; it < NITER; ++it) {
        int* dr = done + (it & 1);
        int* dw = done + ((it + 1) & 1);
        row_kernel<<<N, 256, 0, stream>>>(C, u, v, absd, dr);
        colpart_kernel<<<dim3(8, 64), 256, 0, stream>>>(C, u, v, part, dr);
        colreduce_kernel<<<9, 256, 0, stream>>>(part, v, absd, dr, dw);
    }

    final_part_kernel<<<FINAL_BLOCKS, 256, 0, stream>>>(C, u, v, fpart);
    final_reduce_kernel<<<1, 256, 0, stream>>>(fpart, out);
}